// Decoder_40492951667511
// MI455X (gfx1250) — compile-verified
//
#include <hip/hip_runtime.h>
#include <math.h>

// Sizes from the reference
constexpr int kV = 128000;  // vocab
constexpr int kH = 1024;    // hidden

typedef __attribute__((ext_vector_type(2))) float v2f;
typedef __attribute__((ext_vector_type(8))) float v8f;

// ---------------------------------------------------------------------------
// K1: GRU gate GEMVs:  gi = x @ w_ih.T + b_ih ;  gh = h @ w_hh.T + b_hh
// One wave per output row (3072 rows), both matrices streamed together.
// ---------------------------------------------------------------------------
__global__ __launch_bounds__(256) void gru_gemv(const int* __restrict__ token,
                                                const float* __restrict__ emb,
                                                const float* __restrict__ hidden,
                                                const float* __restrict__ w_ih,
                                                const float* __restrict__ w_hh,
                                                const float* __restrict__ b_ih,
                                                const float* __restrict__ b_hh,
                                                float* __restrict__ gi,
                                                float* __restrict__ gh) {
  const int lane = threadIdx.x & 31;
  const int wave = threadIdx.x >> 5;
  const int row  = blockIdx.x * 8 + wave;   // grid = 384 blocks -> 3072 rows
  if (row >= 3 * kH) return;

  const float* x  = emb + (size_t)token[0] * kH;
  const float* wi = w_ih + (size_t)row * kH;
  const float* wh = w_hh + (size_t)row * kH;

  float si = 0.0f, sh = 0.0f;
#pragma unroll
  for (int k = lane * 4; k < kH; k += 32 * 4) {
    float4 xv  = *reinterpret_cast<const float4*>(x + k);
    float4 hv  = *reinterpret_cast<const float4*>(hidden + k);
    float4 wiv = *reinterpret_cast<const float4*>(wi + k);
    float4 whv = *reinterpret_cast<const float4*>(wh + k);
    si += xv.x * wiv.x + xv.y * wiv.y + xv.z * wiv.z + xv.w * wiv.w;
    sh += hv.x * whv.x + hv.y * whv.y + hv.z * whv.z + hv.w * whv.w;
  }
  // wave32 reduction
  for (int off = 16; off > 0; off >>= 1) {
    si += __shfl_down(si, off, 32);
    sh += __shfl_down(sh, off, 32);
  }
  if (lane == 0) {
    gi[row] = si + b_ih[row];
    gh[row] = sh + b_hh[row];
  }
}

// ---------------------------------------------------------------------------
// K2: fused GRU gates -> h_new (also written to the output tail)
// ---------------------------------------------------------------------------
__global__ void gru_gates(const float* __restrict__ gi, const float* __restrict__ gh,
                          const float* __restrict__ hidden,
                          float* __restrict__ h_new, float* __restrict__ out_tail) {
  int i = blockIdx.x * blockDim.x + threadIdx.x;
  if (i >= kH) return;
  float ir = gi[i],          hr = gh[i];
  float iz = gi[kH + i],     hz = gh[kH + i];
  float in_ = gi[2 * kH + i], hn = gh[2 * kH + i];
  float r = 1.0f / (1.0f + expf(-(ir + hr)));
  float z = 1.0f / (1.0f + expf(-(iz + hz)));
  float n = tanhf(in_ + r * hn);
  float hv = (1.0f - z) * n + z * hidden[i];
  h_new[i]    = hv;
  out_tail[i] = hv;
}

// ---------------------------------------------------------------------------
// K3: logits = h_new @ w_out.T + b_out via V_WMMA_F32_16X16X4_F32.
// One wave -> 16 logits. A holds h in row M=0 (lane0: K0,K1; lane16: K2,K3;
// other lanes zero). B[k][n] = w_out[j0+n][k0+k]. 256 accumulating WMMAs.
// EXEC stays all-ones across the WMMA loop (grid is exact, masking via select).
// ---------------------------------------------------------------------------
__global__ __launch_bounds__(256) void logits_wmma(const float* __restrict__ w_out,
                                                   const float* __restrict__ b_out,
                                                   const float* __restrict__ h_new,
                                                   float* __restrict__ logits) {
  __shared__ float sh[kH];
  for (int i = threadIdx.x; i < kH; i += blockDim.x) sh[i] = h_new[i];
  __syncthreads();

  const int lane  = threadIdx.x & 31;
  const int wave  = threadIdx.x >> 5;
  const int tile  = blockIdx.x * 8 + wave;   // grid = 1000 blocks -> 8000 tiles
  const int j0    = tile * 16;
  const int n     = lane & 15;               // output column within tile
  const int khalf = lane >> 4;               // 0: K0/K1, 1: K2/K3
  const bool row0 = (n == 0);                // lanes that carry A row M=0

  const float* wrow = w_out + (size_t)(j0 + n) * kH + 2 * khalf;

  v8f c = {};
#pragma unroll 4
  for (int k0 = 0; k0 < kH; k0 += 4) {
    float2 av = *reinterpret_cast<const float2*>(&sh[k0 + 2 * khalf]);  // ds_load_b64
    v2f a;
    a.x = row0 ? av.x : 0.0f;
    a.y = row0 ? av.y : 0.0f;
    float2 bv = *reinterpret_cast<const float2*>(wrow + k0);            // global b64
    v2f b;
    b.x = bv.x;
    b.y = bv.y;
    c = __builtin_amdgcn_wmma_f32_16x16x4_f32(false, a, false, b,
                                              (short)0, c, false, false);
  }
  // D row M=0 lives in c[0] on lanes 0..15 (N = lane)
  if (lane < 16) {
    logits[j0 + lane] = c[0] + b_out[j0 + lane];
  }
}

// ---------------------------------------------------------------------------
// K4/K5/K6: online log-softmax (per-block (max, sumexp) -> combine -> apply)
// ---------------------------------------------------------------------------
__global__ __launch_bounds__(256) void softmax_partial(const float* __restrict__ logits,
                                                       float* __restrict__ pm,
                                                       float* __restrict__ ps) {
  __shared__ float smx[256];
  __shared__ float ssum[256];
  const int tid  = threadIdx.x;
  const int base = blockIdx.x * 1024;        // grid = 125 blocks, kV = 125*1024
  float m = -INFINITY, s = 0.0f;
#pragma unroll
  for (int k = 0; k < 4; ++k) {
    float x = logits[base + k * 256 + tid];
    if (x > m) { s = s * expf(m - x) + 1.0f; m = x; }
    else       { s += expf(x - m); }
  }
  smx[tid] = m; ssum[tid] = s;
  __syncthreads();
  for (int off = 128; off > 0; off >>= 1) {
    if (tid < off) {
      float m2 = smx[tid + off], s2 = ssum[tid + off];
      float M  = fmaxf(smx[tid], m2);
      ssum[tid] = ssum[tid] * expf(smx[tid] - M) + s2 * expf(m2 - M);
      smx[tid]  = M;
    }
    __syncthreads();
  }
  if (tid == 0) { pm[blockIdx.x] = smx[0]; ps[blockIdx.x] = ssum[0]; }
}

__global__ __launch_bounds__(128) void softmax_final(const float* __restrict__ pm,
                                                     const float* __restrict__ ps,
                                                     float* __restrict__ logz, int nblk) {
  __shared__ float smx[128];
  __shared__ float ssum[128];
  const int tid = threadIdx.x;
  float m = -INFINITY, s = 0.0f;
  if (tid < nblk) { m = pm[tid]; s = ps[tid]; }
  smx[tid] = m; ssum[tid] = s;
  __syncthreads();
  for (int off = 64; off > 0; off >>= 1) {
    if (tid < off) {
      float m2 = smx[tid + off], s2 = ssum[tid + off];
      float M  = fmaxf(smx[tid], m2);
      ssum[tid] = ssum[tid] * expf(smx[tid] - M) + s2 * expf(m2 - M);
      smx[tid]  = M;
    }
    __syncthreads();
  }
  if (tid == 0) logz[0] = smx[0] + logf(ssum[0]);
}

__global__ void finalize(const float* __restrict__ logits,
                         const float* __restrict__ logz,
                         float* __restrict__ out) {
  int j = blockIdx.x * blockDim.x + threadIdx.x;
  if (j < kV) out[j] = logits[j] - logz[0];
}

// ---------------------------------------------------------------------------
// Launcher. Input order: token, hidden, emb, w_ih, w_hh, b_ih, b_hh, w_out, b_out
// d_out = [log_probs (128000 f32) | h_new (1024 f32)]
// ---------------------------------------------------------------------------
extern "C" void kernel_launch(void* const* d_in, const int* in_sizes, int n_in,
                              void* d_out, int out_size, void* d_ws, size_t ws_size,
                              hipStream_t stream) {
  const int*   token  = (const int*)d_in[0];
  const float* hidden = (const float*)d_in[1];
  const float* emb    = (const float*)d_in[2];
  const float* w_ih   = (const float*)d_in[3];
  const float* w_hh   = (const float*)d_in[4];
  const float* b_ih   = (const float*)d_in[5];
  const float* b_hh   = (const float*)d_in[6];
  const float* w_out  = (const float*)d_in[7];
  const float* b_out  = (const float*)d_in[8];
  float*       out    = (float*)d_out;

  float* ws     = (float*)d_ws;
  float* logits = ws;                  // kV
  float* gi     = logits + kV;         // 3*kH
  float* gh     = gi + 3 * kH;         // 3*kH
  float* hn     = gh + 3 * kH;         // kH
  float* pm     = hn + kH;             // 128
  float* ps     = pm + 128;            // 128
  float* lz     = ps + 128;            // 1

  gru_gemv<<<384, 256, 0, stream>>>(token, emb, hidden, w_ih, w_hh, b_ih, b_hh, gi, gh);
  gru_gates<<<4, 256, 0, stream>>>(gi, gh, hidden, hn, out + kV);
  logits_wmma<<<1000, 256, 0, stream>>>(w_out, b_out, hn, logits);
  softmax_partial<<<125, 256, 0, stream>>>(logits, pm, ps);
  softmax_final<<<1, 128, 0, stream>>>(pm, ps, lz, 125);
  finalize<<<500, 256, 0, stream>>>(logits, lz, out);
}